// PhysicalLayer_8796093022718
// MI455X (gfx1250) — compile-verified
//
#include <hip/hip_runtime.h>
#include <math.h>
#include <stdint.h>

typedef float v2f __attribute__((ext_vector_type(2)));
typedef float v8f __attribute__((ext_vector_type(8)));

#define HW      65536      // 256*256
#define NIMG    128        // 4*32
#define IMGSZ   8388608    // 128*65536 floats

// workspace layout (float offsets)
#define OFF_DR    0
#define OFF_DI    65536
#define OFF_COSM  131072
#define OFF_SINM  196608
#define OFF_KERN  262144   // 32*81 used
#define OFF_SCALE 266240   // 128 used
#define OFF_PART  266496   // 128*32 = 4096 used
#define OFF_XR    274688
#define OFF_XI    (OFF_XR + IMGSZ)
#define OFF_TR    (OFF_XI + IMGSZ)
#define OFF_TI    (OFF_TR + IMGSZ)
// intens aliases OFF_XR (Xr dead after GEMM1)

// ---- CDNA5 async global->LDS staging (ASYNCcnt path, per CDNA5 ISA) ----
__device__ __forceinline__ void async_ld_b32(void* lds, const void* g) {
  asm volatile("global_load_async_to_lds_b32 %0, %1, off"
               :: "v"((unsigned)(uintptr_t)lds), "v"(g) : "memory");
}
__device__ __forceinline__ void async_ld_b64(void* lds, const void* g) {
  asm volatile("global_load_async_to_lds_b64 %0, %1, off"
               :: "v"((unsigned)(uintptr_t)lds), "v"(g) : "memory");
}
__device__ __forceinline__ void wait_async0() {
#if __has_builtin(__builtin_amdgcn_s_wait_asynccnt)
  __builtin_amdgcn_s_wait_asynccnt(0);
#else
  asm volatile("s_wait_asynccnt 0x0" ::: "memory");
#endif
}

// ---------------- Stage 0: DFT matrix, mask cos/sin, blur kernels ----------------
__global__ __launch_bounds__(256) void precompute_kernel(
    const float* __restrict__ mask, const float* __restrict__ stds_u,
    const float* __restrict__ gauss,
    float* __restrict__ Dr, float* __restrict__ Di,
    float* __restrict__ cosm, float* __restrict__ sinm,
    float* __restrict__ kern) {
  int p = blockIdx.x * 256 + threadIdx.x;        // 0..65535
  int k = p >> 8, n = p & 255;
  // centered ortho DFT: D[k,n] = exp(-2*pi*i*(k-128)*(n-128)/256) / 16
  int m = ((k - 128) * (n - 128)) & 255;         // exact mod-256 phase
  float th = -6.28318530717958647f * (float)m * (1.0f / 256.0f);
  float s, c;
  sincosf(th, &s, &c);
  Dr[p] = c * 0.0625f;
  Di[p] = s * 0.0625f;
  float mv = mask[p];
  float sm, cm;
  sincosf(mv, &sm, &cm);
  cosm[p] = cm;
  sinm[p] = sm;
  if (p < 32 * 81) {
    int e = p / 81, t = p - e * 81;
    float std = 0.5f + 3.0f * stds_u[e];
    float iv = 1.0f / (std * std);
    kern[p] = iv * 0.15915494309189535f * powf(gauss[t], iv);
  }
}

// ---------------- Stage 1: field = exp(i*mask) * emitter ----------------
__global__ __launch_bounds__(256) void field_kernel(
    const float* __restrict__ pe, const float* __restrict__ cosm,
    const float* __restrict__ sinm, float* __restrict__ Xr,
    float* __restrict__ Xi) {
  int p = blockIdx.x * 256 + threadIdx.x;
  size_t i = (size_t)blockIdx.y * HW + (size_t)p;
  float er = pe[2 * i], ei = pe[2 * i + 1];
  float c = cosm[p], s = sinm[p];
  Xr[i] = c * er - s * ei;
  Xi[i] = c * ei + s * er;
}

// ---------------- Stage 2: T = D * X (complex, per image), fp32 WMMA ----------------
__global__ __launch_bounds__(256) void dft_rows_kernel(
    const float* __restrict__ Dr, const float* __restrict__ Di,
    const float* __restrict__ Xr, const float* __restrict__ Xi,
    float* __restrict__ Tr, float* __restrict__ Ti) {
  __shared__ float sAr[64 * 18], sAi[64 * 18];   // A tile [64][16] stride-18
  __shared__ float sBr[32 * 18], sBi[32 * 18];   // B tile transposed [n=32][k=16]
  const int tid = threadIdx.x;
  const int lane = tid & 31;
  const int wave = tid >> 5;
  const int img = blockIdx.z;
  const int m0g = blockIdx.y * 64;
  const int n0g = blockIdx.x * 32;
  const size_t ib = (size_t)img * HW;
  const int laneLo = lane & 15;
  const int hi = lane >> 4;
  const int m0w = (wave & 3) * 16;
  const int n0w = (wave >> 2) * 16;

  v8f accRR = {}, accII = {}, accRI = {}, accIR = {};

  for (int k0 = 0; k0 < 256; k0 += 16) {
    __syncthreads();
    {  // A = D rows m0g..+63, cols k0..+15 : async B64 (contiguous k-pair per lane)
      int c2 = (tid & 7) * 2;
      int r0 = tid >> 3;                        // 32 rows per pass
      for (int rr = r0; rr < 64; rr += 32) {
        int g = (m0g + rr) * 256 + k0 + c2;
        async_ld_b64(&sAr[rr * 18 + c2], &Dr[g]);
        async_ld_b64(&sAi[rr * 18 + c2], &Di[g]);
      }
    }
    {  // B = X rows k0..+15, cols n0g..+31 : async B32 (transposed into LDS)
      int c = tid & 31, r0 = tid >> 5;          // 8 rows per pass
      for (int rr = r0; rr < 16; rr += 8) {
        size_t g = ib + (size_t)(k0 + rr) * 256 + n0g + c;
        async_ld_b32(&sBr[c * 18 + rr], &Xr[g]);
        async_ld_b32(&sBi[c * 18 + rr], &Xi[g]);
      }
    }
    wait_async0();
    __syncthreads();
#pragma unroll
    for (int kk = 0; kk < 16; kk += 4) {
      int aoff = (m0w + laneLo) * 18 + kk + 2 * hi;
      int boff = (n0w + laneLo) * 18 + kk + 2 * hi;
      v2f ar = *(const v2f*)&sAr[aoff];
      v2f ai = *(const v2f*)&sAi[aoff];
      v2f br = *(const v2f*)&sBr[boff];
      v2f bi = *(const v2f*)&sBi[boff];
      accRR = __builtin_amdgcn_wmma_f32_16x16x4_f32(false, ar, false, br, (short)0, accRR, false, false);
      accII = __builtin_amdgcn_wmma_f32_16x16x4_f32(false, ai, false, bi, (short)0, accII, false, false);
      accRI = __builtin_amdgcn_wmma_f32_16x16x4_f32(false, ar, false, bi, (short)0, accRI, false, false);
      accIR = __builtin_amdgcn_wmma_f32_16x16x4_f32(false, ai, false, br, (short)0, accIR, false, false);
    }
  }
  v8f tr = accRR - accII;
  v8f ti = accRI + accIR;
  int col = n0g + n0w + laneLo;
#pragma unroll
  for (int r = 0; r < 8; ++r) {
    int row = m0g + m0w + r + hi * 8;
    Tr[ib + (size_t)row * 256 + col] = tr[r];
    Ti[ib + (size_t)row * 256 + col] = ti[r];
  }
}

// ---------------- Stage 3: intens = |T * D|^2 + deterministic partial sums ----------------
__global__ __launch_bounds__(256) void dft_cols_kernel(
    const float* __restrict__ Tr, const float* __restrict__ Ti,
    const float* __restrict__ Dr, const float* __restrict__ Di,
    float* __restrict__ intens, float* __restrict__ partials) {
  __shared__ float sAr[64 * 18], sAi[64 * 18];
  __shared__ float sBr[32 * 18], sBi[32 * 18];
  __shared__ float sPart[8];
  const int tid = threadIdx.x;
  const int lane = tid & 31;
  const int wave = tid >> 5;
  const int img = blockIdx.z;
  const int m0g = blockIdx.y * 64;
  const int n0g = blockIdx.x * 32;
  const size_t ib = (size_t)img * HW;
  const int laneLo = lane & 15;
  const int hi = lane >> 4;
  const int m0w = (wave & 3) * 16;
  const int n0w = (wave >> 2) * 16;

  v8f accRR = {}, accII = {}, accRI = {}, accIR = {};

  for (int k0 = 0; k0 < 256; k0 += 16) {
    __syncthreads();
    {  // A = T rows m0g..+63, cols k0..+15 (per image) : async B64
      int c2 = (tid & 7) * 2;
      int r0 = tid >> 3;
      for (int rr = r0; rr < 64; rr += 32) {
        size_t g = ib + (size_t)(m0g + rr) * 256 + k0 + c2;
        async_ld_b64(&sAr[rr * 18 + c2], &Tr[g]);
        async_ld_b64(&sAi[rr * 18 + c2], &Ti[g]);
      }
    }
    {  // B = D rows k0..+15, cols n0g..+31 : async B32 (transposed)
      int c = tid & 31, r0 = tid >> 5;
      for (int rr = r0; rr < 16; rr += 8) {
        int g = (k0 + rr) * 256 + n0g + c;
        async_ld_b32(&sBr[c * 18 + rr], &Dr[g]);
        async_ld_b32(&sBi[c * 18 + rr], &Di[g]);
      }
    }
    wait_async0();
    __syncthreads();
#pragma unroll
    for (int kk = 0; kk < 16; kk += 4) {
      int aoff = (m0w + laneLo) * 18 + kk + 2 * hi;
      int boff = (n0w + laneLo) * 18 + kk + 2 * hi;
      v2f ar = *(const v2f*)&sAr[aoff];
      v2f ai = *(const v2f*)&sAi[aoff];
      v2f br = *(const v2f*)&sBr[boff];
      v2f bi = *(const v2f*)&sBi[boff];
      accRR = __builtin_amdgcn_wmma_f32_16x16x4_f32(false, ar, false, br, (short)0, accRR, false, false);
      accII = __builtin_amdgcn_wmma_f32_16x16x4_f32(false, ai, false, bi, (short)0, accII, false, false);
      accRI = __builtin_amdgcn_wmma_f32_16x16x4_f32(false, ar, false, bi, (short)0, accRI, false, false);
      accIR = __builtin_amdgcn_wmma_f32_16x16x4_f32(false, ai, false, br, (short)0, accIR, false, false);
    }
  }
  v8f yr = accRR - accII;
  v8f yi = accRI + accIR;
  int col = n0g + n0w + laneLo;
  float ps = 0.0f;
#pragma unroll
  for (int r = 0; r < 8; ++r) {
    int row = m0g + m0w + r + hi * 8;
    float v = yr[r] * yr[r] + yi[r] * yi[r];
    intens[ib + (size_t)row * 256 + col] = v;
    ps += v;
  }
  for (int off = 16; off > 0; off >>= 1) ps += __shfl_xor(ps, off, 32);
  if (lane == 0) sPart[wave] = ps;
  __syncthreads();
  if (tid == 0) {
    float t = 0.0f;
    for (int w = 0; w < 8; ++w) t += sPart[w];
    partials[(size_t)img * 32 + blockIdx.y * 8 + blockIdx.x] = t;
  }
}

// ---------------- Stage 4: per-image normalization scale ----------------
__global__ void scale_kernel(const float* __restrict__ nph,
                             const float* __restrict__ partials,
                             float* __restrict__ scale) {
  int t = threadIdx.x;  // 0..127 == (b*32+e)
  float s = 0.0f;
  for (int i = 0; i < 32; ++i) s += partials[t * 32 + i];
  scale[t] = nph[t] / s;
}

// ---------------- Stage 5: 9x9 per-emitter blur, sum over emitters, crop ----------------
__global__ __launch_bounds__(256) void conv_crop_kernel(
    const float* __restrict__ intens, const float* __restrict__ kern,
    const float* __restrict__ scale, float* __restrict__ out) {
  __shared__ float sK[32 * 81];
  __shared__ float sT[24 * 25];
  const int tid = threadIdx.x;
  const int b = blockIdx.z;
  const int lx = tid & 15, ly = tid >> 4;
  const int ox = blockIdx.x * 16 + lx, oy = blockIdx.y * 16 + ly;
  for (int i = tid; i < 32 * 81; i += 256) sK[i] = kern[i];
  const int ry0 = blockIdx.y * 16 + 64;   // crop start 66 - pad 2 = 64 (interior)
  const int rx0 = blockIdx.x * 16 + 64;
  float acc = 0.0f;
  for (int e = 0; e < 32; ++e) {
    __syncthreads();
    const float* src = intens + (size_t)(b * 32 + e) * HW;
    for (int i = tid; i < 24 * 24; i += 256) {
      int r = i / 24, c = i - r * 24;
      sT[r * 25 + c] = src[(ry0 + r) * 256 + rx0 + c];
    }
    __syncthreads();
    float a = 0.0f;
#pragma unroll
    for (int i = 0; i < 9; ++i)
#pragma unroll
      for (int j = 0; j < 9; ++j)
        a += sK[e * 81 + i * 9 + j] * sT[(ly + i) * 25 + lx + j];
    acc += scale[b * 32 + e] * a;
  }
  if (ox < 121 && oy < 121)
    out[(size_t)b * 121 * 121 + oy * 121 + ox] = acc;
}

extern "C" void kernel_launch(void* const* d_in, const int* in_sizes, int n_in,
                              void* d_out, int out_size, void* d_ws, size_t ws_size,
                              hipStream_t stream) {
  (void)in_sizes; (void)n_in; (void)out_size; (void)ws_size;
  const float* mask   = (const float*)d_in[0];
  const float* pe     = (const float*)d_in[1];
  const float* nph    = (const float*)d_in[2];
  const float* stds_u = (const float*)d_in[3];
  const float* gauss  = (const float*)d_in[4];
  float* ws = (float*)d_ws;
  float* Dr = ws + OFF_DR;   float* Di = ws + OFF_DI;
  float* cosm = ws + OFF_COSM; float* sinm = ws + OFF_SINM;
  float* kern = ws + OFF_KERN; float* scl = ws + OFF_SCALE;
  float* part = ws + OFF_PART;
  float* Xr = ws + OFF_XR;   float* Xi = ws + OFF_XI;
  float* Tr = ws + OFF_TR;   float* Ti = ws + OFF_TI;
  float* intens = Xr;  // Xr is dead after dft_rows_kernel

  precompute_kernel<<<256, 256, 0, stream>>>(mask, stds_u, gauss, Dr, Di, cosm, sinm, kern);
  field_kernel<<<dim3(256, NIMG), 256, 0, stream>>>(pe, cosm, sinm, Xr, Xi);
  dft_rows_kernel<<<dim3(8, 4, NIMG), 256, 0, stream>>>(Dr, Di, Xr, Xi, Tr, Ti);
  dft_cols_kernel<<<dim3(8, 4, NIMG), 256, 0, stream>>>(Tr, Ti, Dr, Di, intens, part);
  scale_kernel<<<1, 128, 0, stream>>>(nph, part, scl);
  conv_crop_kernel<<<dim3(8, 8, 4), 256, 0, stream>>>(intens, kern, scl, (float*)d_out);
}